// LocalRefinementBlock_40200893891375
// MI455X (gfx1250) — compile-verified
//
#include <hip/hip_runtime.h>
#include <hip/hip_bf16.h>

// ---------------------------------------------------------------------------
// LocalRefinementBlock for MI455X (gfx1250, wave32).
// Pipeline: build_wnode -> knn -> edge_msg (gather+message+aggregate) ->
//           node_update GEMM via V_WMMA_F32_16X16X4_F32 (fp32 WMMA).
// ---------------------------------------------------------------------------

typedef __attribute__((ext_vector_type(2))) float v2f;
typedef __attribute__((ext_vector_type(8))) float v8f;

#define NB 4
#define NN 4096
#define NNODE (NB * NN)
#define KNB 8
#define DF 240          // 64 + 3*32 + 5*16

// Wigner coupling tensor M[k][i][j] (real basis, symmetric in i,j), /sqrt(5)
// applied via INV_SQRT5 at use sites.
__constant__ float Mw[5][3][3] = {
    {{0.f, 0.70710678118654752f, 0.f},
     {0.70710678118654752f, 0.f, 0.f},
     {0.f, 0.f, 0.f}},
    {{0.f, 0.f, 0.70710678118654752f},
     {0.f, 0.f, 0.f},
     {0.70710678118654752f, 0.f, 0.f}},
    {{0.f, 0.f, 0.f},
     {0.f, 0.f, 0.70710678118654752f},
     {0.f, 0.70710678118654752f, 0.f}},
    {{0.70710678118654752f, 0.f, 0.f},
     {0.f, -0.70710678118654752f, 0.f},
     {0.f, 0.f, 0.f}},
    {{-0.40824829046386302f, 0.f, 0.f},
     {0.f, -0.40824829046386302f, 0.f},
     {0.f, 0.f, 0.81649658092772603f}}};

#define INV_SQRT5 0.44721359549995794f
#define C1 0.11180339887498948f   // sqrt(1/80)
#define C4 0.19364916731037085f   // sqrt(3/80)
#define C2 0.10206207261596577f   // sqrt(1/96)
#define C3 0.39528470752104744f   // sqrt(5/32)

// ---------------------------------------------------------------------------
// Kernel 1: build the 240x240 block-diagonal node-update weight matrix.
// Wnode[in][out]; out = agg @ Wnode.
// ---------------------------------------------------------------------------
__global__ void build_wnode_kernel(const float* __restrict__ L0,
                                   const float* __restrict__ L1,
                                   const float* __restrict__ L2,
                                   float* __restrict__ wnode) {
  int idx = blockIdx.x * blockDim.x + threadIdx.x;
  if (idx >= DF * DF) return;
  int i = idx / DF;
  int o = idx % DF;
  float v = 0.f;
  if (i < 64 && o < 64) {
    v = L0[i * 64 + o] * 0.125f;                       // 1/sqrt(64)
  } else if (i >= 64 && i < 160 && o >= 64 && o < 160) {
    int a = i - 64, c = o - 64;
    int u = a / 3, ki = a % 3, w = c / 3, ko = c % 3;
    if (ki == ko) v = L1[u * 32 + w] * 0.17677669529663687f;  // 1/sqrt(32)
  } else if (i >= 160 && o >= 160) {
    int a = i - 160, c = o - 160;
    int u = a / 5, ki = a % 5, w = c / 5, ko = c % 5;
    if (ki == ko) v = L2[u * 16 + w] * 0.25f;          // 1/sqrt(16)
  }
  wnode[idx] = v;
}

// ---------------------------------------------------------------------------
// Kernel 2: brute-force per-batch kNN (K=8), LDS-tiled. Strict-< insertion
// sort reproduces top_k stable tie-breaking (earliest index wins on ties).
// ---------------------------------------------------------------------------
__global__ __launch_bounds__(128) void knn_kernel(const float* __restrict__ coords,
                                                  int* __restrict__ nbr) {
  __shared__ float sx[128], sy[128], sz[128];
  const int tid = threadIdx.x;
  const int bpb = NN / 128;                 // blocks per batch
  const int b = blockIdx.x / bpb;
  const int i = (blockIdx.x % bpb) * 128 + tid;
  const float* cb = coords + (size_t)b * NN * 3;
  const float px = cb[i * 3 + 0];
  const float py = cb[i * 3 + 1];
  const float pz = cb[i * 3 + 2];

  float bd[KNB];
  int bi[KNB];
#pragma unroll
  for (int q = 0; q < KNB; ++q) { bd[q] = 3.0e38f; bi[q] = 0; }

  for (int j0 = 0; j0 < NN; j0 += 128) {
    __syncthreads();
    sx[tid] = cb[(j0 + tid) * 3 + 0];
    sy[tid] = cb[(j0 + tid) * 3 + 1];
    sz[tid] = cb[(j0 + tid) * 3 + 2];
    __syncthreads();
    for (int jj = 0; jj < 128; ++jj) {
      const int j = j0 + jj;
      const float dx = px - sx[jj];
      const float dy = py - sy[jj];
      const float dz = pz - sz[jj];
      float d2 = dx * dx + dy * dy + dz * dz;
      if (j == i) d2 += 1.0e9f;             // exclude self (matches reference)
      if (d2 < bd[KNB - 1]) {
        bd[KNB - 1] = d2;
        bi[KNB - 1] = j;
#pragma unroll
        for (int q = KNB - 1; q > 0; --q) {
          if (bd[q] < bd[q - 1]) {
            float td = bd[q]; bd[q] = bd[q - 1]; bd[q - 1] = td;
            int ti = bi[q]; bi[q] = bi[q - 1]; bi[q - 1] = ti;
          }
        }
      }
    }
  }
  const int base = (b * NN + i) * KNB;
#pragma unroll
  for (int q = 0; q < KNB; ++q) nbr[base + q] = b * NN + bi[q];
}

// ---------------------------------------------------------------------------
// Kernel 3: per-node message computation + aggregation (no atomics: each node
// owns exactly its K incoming edges). One 256-thread block per node; thread c
// (< 240) owns output channel c of agg[n].
// ---------------------------------------------------------------------------
__global__ __launch_bounds__(256) void edge_msg_kernel(
    const float* __restrict__ feats, const float* __restrict__ coords,
    const float* __restrict__ W1, const float* __restrict__ W2,
    const float* __restrict__ W3, const float* __restrict__ W4,
    const int* __restrict__ nbr, float* __restrict__ agg) {
  __shared__ float sW1[64 * 32];   // [u][w]
  __shared__ float sW2[32 * 64];   // [u][w]
  __shared__ float sW3[32 * 16];   // [u][w]
  __shared__ float sW4[16 * 32];   // [u][w]
  __shared__ float xj[DF];
  __shared__ float sev[3];
  __shared__ float s_s[32];        // x1 . ev per u
  __shared__ float s_g1[32];       // x0 @ W1
  __shared__ float s_t[48];        // [u<16][k<3] : x2 contracted with C211.ev
  __shared__ float s_t2[160];      // [u<32][k<5] : x1 contracted with C112.ev

  const int tid = threadIdx.x;
  const int n = blockIdx.x;

  for (int i = tid; i < 2048; i += 256) { sW1[i] = W1[i]; sW2[i] = W2[i]; }
  for (int i = tid; i < 512; i += 256) { sW3[i] = W3[i]; sW4[i] = W4[i]; }

  const float rcx = coords[n * 3 + 0];
  const float rcy = coords[n * 3 + 1];
  const float rcz = coords[n * 3 + 2];

  float acc = 0.f;

  for (int e = 0; e < KNB; ++e) {
    const int snd = nbr[n * KNB + e];
    __syncthreads();  // previous-iteration consumers done before overwrite
    if (tid < DF) xj[tid] = feats[(size_t)snd * DF + tid];
    if (tid < 3) sev[tid] = coords[snd * 3 + tid] -
                            (tid == 0 ? rcx : (tid == 1 ? rcy : rcz));
    __syncthreads();
    const float ev0 = sev[0], ev1 = sev[1], ev2 = sev[2];

    // Stage B: shared intermediates (272 virtual work items).
    for (int w = tid; w < 272; w += 256) {
      if (w < 32) {                                   // s[u] = x1[u,:].ev
        const int u = w;
        s_s[u] = xj[64 + u * 3 + 0] * ev0 + xj[64 + u * 3 + 1] * ev1 +
                 xj[64 + u * 3 + 2] * ev2;
      } else if (w < 64) {                            // g1[w] = sum_u x0[u] W1[u,w]
        const int ww = w - 32;
        float g = 0.f;
        for (int u = 0; u < 64; ++u) g += xj[u] * sW1[u * 32 + ww];
        s_g1[ww] = g;
      } else if (w < 112) {                           // t[u,k] = sum_a x2[u,a] Q[a,k]
        const int idx = w - 64;
        const int u = idx / 3, k = idx % 3;
        float tt = 0.f;
#pragma unroll
        for (int a = 0; a < 5; ++a) {
          const float q = (Mw[a][0][k] * ev0 + Mw[a][1][k] * ev1 +
                           Mw[a][2][k] * ev2) * INV_SQRT5;
          tt += xj[160 + u * 5 + a] * q;
        }
        s_t[idx] = tt;
      } else {                                        // t2[u,k] = sum_i x1[u,i] R[i,k]
        const int idx = w - 112;
        const int u = idx / 5, k = idx % 5;
        const float r0 = (Mw[k][0][0] * ev0 + Mw[k][0][1] * ev1 + Mw[k][0][2] * ev2) * INV_SQRT5;
        const float r1 = (Mw[k][1][0] * ev0 + Mw[k][1][1] * ev1 + Mw[k][1][2] * ev2) * INV_SQRT5;
        const float r2 = (Mw[k][2][0] * ev0 + Mw[k][2][1] * ev1 + Mw[k][2][2] * ev2) * INV_SQRT5;
        s_t2[idx] = xj[64 + u * 3 + 0] * r0 + xj[64 + u * 3 + 1] * r1 +
                    xj[64 + u * 3 + 2] * r2;
      }
    }
    __syncthreads();

    // Stage C: per-channel accumulation.
    if (tid < 64) {                                   // m0[w]
      float m = 0.f;
      for (int u = 0; u < 32; ++u) m += s_s[u] * sW2[u * 64 + tid];
      acc += C2 * m;
    } else if (tid < 160) {                           // m1[w,k]
      const int idx = tid - 64;
      const int w = idx / 3, k = idx % 3;
      float m = 0.f;
      for (int u = 0; u < 16; ++u) m += s_t[u * 3 + k] * sW4[u * 32 + w];
      const float evk = (k == 0) ? ev0 : ((k == 1) ? ev1 : ev2);
      acc += C1 * evk * s_g1[w] + C4 * m;
    } else if (tid < 240) {                           // m2[w,k]
      const int idx = tid - 160;
      const int w = idx / 5, k = idx % 5;
      float m = 0.f;
      for (int u = 0; u < 32; ++u) m += s_t2[u * 5 + k] * sW3[u * 16 + w];
      acc += C3 * m;
    }
  }

  if (tid < DF) agg[(size_t)n * DF + tid] = acc;
}

// ---------------------------------------------------------------------------
// Kernel 4: Y[16384,240] = Agg[16384,240] @ Wnode[240,240] via fp32 WMMA
// (V_WMMA_F32_16X16X4_F32). One wave per 16-row tile; loops 15 N-tiles x 60
// K-chunks. Lane mapping per CDNA5 ISA 7.12.2:
//   A: lane L holds A[L&15][2*(L>>4)] , A[L&15][2*(L>>4)+1]
//   B: lane L holds B[2*(L>>4)][L&15], B[2*(L>>4)+1][L&15]
//   D: VGPR v of lane L holds D[v + 8*(L>>4)][L&15]
// ---------------------------------------------------------------------------
__global__ __launch_bounds__(256) void node_update_wmma_kernel(
    const float* __restrict__ agg, const float* __restrict__ wnode,
    float* __restrict__ out) {
  const int lane = threadIdx.x & 31;
  const int wave = threadIdx.x >> 5;
  const int mtile = blockIdx.x * 8 + wave;
  const int m0 = mtile * 16;
  const int r = lane & 15;
  const int h = lane >> 4;
  const float* arow = agg + (size_t)(m0 + r) * DF + 2 * h;

  for (int nt = 0; nt < 15; ++nt) {
    const int nc = nt * 16 + r;
    v8f c = {0.f, 0.f, 0.f, 0.f, 0.f, 0.f, 0.f, 0.f};
    for (int k = 0; k < DF; k += 4) {
      v2f a;
      a.x = arow[k + 0];
      a.y = arow[k + 1];
      v2f bv;
      bv.x = wnode[(k + 2 * h + 0) * DF + nc];
      bv.y = wnode[(k + 2 * h + 1) * DF + nc];
      c = __builtin_amdgcn_wmma_f32_16x16x4_f32(
          /*neg_a=*/false, a, /*neg_b=*/false, bv,
          /*c_mod=*/(short)0, c, /*reuse_a=*/false, /*reuse_b=*/false);
    }
#pragma unroll
    for (int v = 0; v < 8; ++v) {
      out[(size_t)(m0 + v + 8 * h) * DF + nc] = c[v];
    }
  }
}

// ---------------------------------------------------------------------------
// Host-side launcher. Inputs (dict order): node_feats, coords, W1, W2, W3,
// W4, L0, L1, L2. Workspace layout (floats):
//   [0, 57600)                       Wnode 240x240
//   [57600, 57600+16384*240)         agg
//   then NNODE*KNB ints              nbr
// ---------------------------------------------------------------------------
extern "C" void kernel_launch(void* const* d_in, const int* in_sizes, int n_in,
                              void* d_out, int out_size, void* d_ws,
                              size_t ws_size, hipStream_t stream) {
  const float* feats  = (const float*)d_in[0];
  const float* coords = (const float*)d_in[1];
  const float* W1 = (const float*)d_in[2];
  const float* W2 = (const float*)d_in[3];
  const float* W3 = (const float*)d_in[4];
  const float* W4 = (const float*)d_in[5];
  const float* L0 = (const float*)d_in[6];
  const float* L1 = (const float*)d_in[7];
  const float* L2 = (const float*)d_in[8];
  float* out = (float*)d_out;

  float* ws = (float*)d_ws;
  float* wnode = ws;
  float* agg = ws + (size_t)DF * DF;
  int* nbr = (int*)(ws + (size_t)DF * DF + (size_t)NNODE * DF);

  build_wnode_kernel<<<(DF * DF + 255) / 256, 256, 0, stream>>>(L0, L1, L2,
                                                                wnode);
  knn_kernel<<<NB * (NN / 128), 128, 0, stream>>>(coords, nbr);
  edge_msg_kernel<<<NNODE, 256, 0, stream>>>(feats, coords, W1, W2, W3, W4,
                                             nbr, agg);
  node_update_wmma_kernel<<<NNODE / (16 * 8), 256, 0, stream>>>(agg, wnode,
                                                                out);
}